// DeformableDetrTransformerEncoderLayer_13469017440318
// MI455X (gfx1250) — compile-verified
//
#include <hip/hip_runtime.h>
#include <math.h>

// ---------------- problem constants (match reference module) ----------------
constexpr int Bc = 2, Dc = 256, NHc = 8, HDc = 32, Lc = 4, Pc = 4, FFc = 1024;
constexpr int Sc = 13294;                       // 100*100 + 50*50 + 25*25 + 13*13
constexpr long long MROWS = (long long)Bc * Sc; // 26588
constexpr float EPSc = 1e-5f;

__device__ __constant__ int d_H[Lc]  = {100, 50, 25, 13};
__device__ __constant__ int d_W[Lc]  = {100, 50, 25, 13};
__device__ __constant__ int d_S0[Lc] = {0, 10000, 12500, 13125};

// ---------------- WMMA types ----------------
typedef __attribute__((ext_vector_type(2))) float v2f;
typedef __attribute__((ext_vector_type(8))) float v8f;
typedef __attribute__((ext_vector_type(4))) int   v4i;

// ---------------- address-space helpers for async-to-LDS ----------------
#define AS1 __attribute__((address_space(1)))
#define AS3 __attribute__((address_space(3)))
// Generic LDS pointers are {SHARED_BASE, offset}; hardware uses addr[31:0] (ISA 10.2),
// so truncation gives the 32-bit LDS address.
__device__ __forceinline__ AS3 v4i* to_lds_v4(const void* p) {
    return (AS3 v4i*)(unsigned int)(unsigned long long)p;
}
// Builtin wants a (non-const) global-AS int4* per the clang diagnostic.
__device__ __forceinline__ AS1 v4i* to_glob_v4(const void* p) {
    return (AS1 v4i*)(unsigned long long)p;
}

#if __has_builtin(__builtin_amdgcn_global_load_async_to_lds_b128) && \
    __has_builtin(__builtin_amdgcn_s_wait_asynccnt)
#define USE_ASYNC_LDS 1
#endif

// =================== tiled f32 WMMA GEMM:  C = A(MxK) * B(KxN) + bias ===================
// block: 256 threads (8 waves). Tile: 128(M) x 64(N), K chunked by 32.
// wave (wm 0..3, wn 0..1) computes a 32x32 block = 2x2 WMMA 16x16 subtiles
// (2 A-frags + 2 B-frags shared across 4 v_wmma_f32_16x16x4_f32 per K-step).
#define MT 128
#define NT 64
#define KT 32
#define APAD 36   // row stride for As: 16B-aligned, conflict-free
#define BPAD 68   // row stride for Bs: 16B-aligned

__global__ __launch_bounds__(256) void gemm_wmma_f32(
    const float* __restrict__ A, const float* __restrict__ Bm,
    const float* __restrict__ bias, float* __restrict__ C,
    int M, int N, int K, int relu)
{
    __shared__ float As[MT][APAD];
    __shared__ float Bs[KT][BPAD];

    const int tid  = threadIdx.x;
    const int lane = tid & 31;
    const int wave = tid >> 5;
    const int wm = wave & 3;          // 0..3 -> M
    const int wn = wave >> 2;         // 0..1 -> N
    const int row0 = blockIdx.x * MT;
    const int col0 = blockIdx.y * NT; // N always a multiple of 64 here

    v8f acc00 = {}, acc01 = {}, acc10 = {}, acc11 = {};

    const int lm    = lane & 15;
    const int koff  = (lane < 16) ? 0 : 2;
    const int mrow0 = wm * 32 + lm;
    const int ncol0 = wn * 32 + lm;

    for (int k0 = 0; k0 < K; k0 += KT) {
#ifdef USE_ASYNC_LDS
        // ---- async tile staging (ASYNCcnt), 16B per lane per issue ----
        for (int i = tid; i < (MT * KT) / 4; i += 256) {
            int r = i >> 3, c = (i & 7) * 4;
            int gr = row0 + r;
            if (gr < M)   // OOB A rows left stale: they only feed D rows >= M (never stored)
                __builtin_amdgcn_global_load_async_to_lds_b128(
                    to_glob_v4(A + (size_t)gr * K + (k0 + c)),
                    to_lds_v4(&As[r][c]), 0, 0);
        }
        for (int i = tid; i < (KT * NT) / 4; i += 256) {
            int r = i >> 4, c = (i & 15) * 4;
            __builtin_amdgcn_global_load_async_to_lds_b128(
                to_glob_v4(Bm + (size_t)(k0 + r) * N + (col0 + c)),
                to_lds_v4(&Bs[r][c]), 0, 0);
        }
        __builtin_amdgcn_s_wait_asynccnt(0);
#else
        // ---- synchronous float4 staging fallback ----
        for (int i = tid; i < (MT * KT) / 4; i += 256) {
            int r = i >> 3, c = (i & 7) * 4;
            int gr = row0 + r;
            float4 val = make_float4(0.f, 0.f, 0.f, 0.f);
            if (gr < M) val = *(const float4*)(A + (size_t)gr * K + (k0 + c));
            *(float4*)(&As[r][c]) = val;
        }
        for (int i = tid; i < (KT * NT) / 4; i += 256) {
            int r = i >> 4, c = (i & 15) * 4;
            *(float4*)(&Bs[r][c]) =
                *(const float4*)(Bm + (size_t)(k0 + r) * N + (col0 + c));
        }
#endif
        __syncthreads();

        #pragma unroll
        for (int kk = 0; kk < KT; kk += 4) {
            v2f a0, a1, b0, b1;
            a0.x = As[mrow0][kk + koff];      a0.y = As[mrow0][kk + koff + 1];
            a1.x = As[mrow0 + 16][kk + koff]; a1.y = As[mrow0 + 16][kk + koff + 1];
            b0.x = Bs[kk + koff][ncol0];      b0.y = Bs[kk + koff + 1][ncol0];
            b1.x = Bs[kk + koff][ncol0 + 16]; b1.y = Bs[kk + koff + 1][ncol0 + 16];
            acc00 = __builtin_amdgcn_wmma_f32_16x16x4_f32(false, a0, false, b0, (short)0, acc00, false, false);
            acc01 = __builtin_amdgcn_wmma_f32_16x16x4_f32(false, a0, false, b1, (short)0, acc01, false, false);
            acc10 = __builtin_amdgcn_wmma_f32_16x16x4_f32(false, a1, false, b0, (short)0, acc10, false, false);
            acc11 = __builtin_amdgcn_wmma_f32_16x16x4_f32(false, a1, false, b1, (short)0, acc11, false, false);
        }
        __syncthreads();
    }

    // ---- epilogue: bias (+ optional relu), predicated on M edge ----
    const int gcol0 = col0 + wn * 32 + lm;
    const float bv0 = bias ? bias[gcol0] : 0.0f;
    const float bv1 = bias ? bias[gcol0 + 16] : 0.0f;
    const int rbase = row0 + wm * 32 + ((lane < 16) ? 0 : 8);
    #pragma unroll
    for (int r = 0; r < 8; ++r) {
        int g0 = rbase + r;
        if (g0 < M) {
            float v0 = acc00[r] + bv0, v1 = acc01[r] + bv1;
            if (relu) { v0 = fmaxf(v0, 0.f); v1 = fmaxf(v1, 0.f); }
            C[(size_t)g0 * N + gcol0]      = v0;
            C[(size_t)g0 * N + gcol0 + 16] = v1;
        }
        int g1 = rbase + 16 + r;
        if (g1 < M) {
            float v0 = acc10[r] + bv0, v1 = acc11[r] + bv1;
            if (relu) { v0 = fmaxf(v0, 0.f); v1 = fmaxf(v1, 0.f); }
            C[(size_t)g1 * N + gcol0]      = v0;
            C[(size_t)g1 * N + gcol0 + 16] = v1;
        }
    }
}

// =================== reference points (B,S,2) ===================
__global__ void ref_points_kernel(const float* __restrict__ valid_ratios, // (B,L,2)
                                  float* __restrict__ ref)                // (B,S,2)
{
    long long idx = (long long)blockIdx.x * blockDim.x + threadIdx.x;
    long long total = (long long)Bc * Sc;
    if (idx >= total) return;
    int s = (int)(idx % Sc);
    int b = (int)(idx / Sc);
    int lvl = Lc - 1;
    #pragma unroll
    for (int l = 0; l < Lc - 1; ++l)
        if (s < d_S0[l + 1]) { lvl = l; break; }
    int W = d_W[lvl], H = d_H[lvl];
    int loc = s - d_S0[lvl];
    int i = loc / W, j = loc % W;
    float vx = valid_ratios[(b * Lc + lvl) * 2 + 0];
    float vy = valid_ratios[(b * Lc + lvl) * 2 + 1];
    ref[idx * 2 + 0] = ((float)j + 0.5f) / (vx * (float)W);
    ref[idx * 2 + 1] = ((float)i + 0.5f) / (vy * (float)H);
}

// =================== q = query + query_pos ===================
__global__ void add_kernel(const float* __restrict__ a, const float* __restrict__ b,
                           float* __restrict__ o, long long n)
{
    long long i = (long long)blockIdx.x * blockDim.x + threadIdx.x;
    if (i < n) o[i] = a[i] + b[i];
}

// =================== softmax over 16 (L*P) per (b,s,h), in place ===================
__global__ void softmax16_kernel(float* __restrict__ p, long long total)
{
    long long i = (long long)blockIdx.x * blockDim.x + threadIdx.x;
    if (i >= total) return;
    float* q = p + i * 16;
    float m = -INFINITY;
    #pragma unroll
    for (int j = 0; j < 16; ++j) m = fmaxf(m, q[j]);
    float e[16], sum = 0.0f;
    #pragma unroll
    for (int j = 0; j < 16; ++j) { e[j] = __expf(q[j] - m); sum += e[j]; }
    float inv = 1.0f / sum;
    #pragma unroll
    for (int j = 0; j < 16; ++j) q[j] = e[j] * inv;
}

// =================== deformable aggregation ===================
// one wave per (b,s,head); lane = head channel (HD == 32 == wave32).
__global__ __launch_bounds__(256) void deform_attn_kernel(
    const float* __restrict__ v,    // (B,S,D) head-major channels
    const float* __restrict__ off,  // (B,S,NH,L,P,2)
    const float* __restrict__ aw,   // (B,S,NH,L*P) softmaxed
    const float* __restrict__ ref,  // (B,S,2)
    float* __restrict__ out)        // (B,S,D)
{
    long long gw = ((long long)blockIdx.x * blockDim.x + threadIdx.x) >> 5;
    const int lane = threadIdx.x & 31;
    const long long total = (long long)Bc * Sc * NHc;
    if (gw >= total) return;
    int h  = (int)(gw % NHc);
    long long bs = gw / NHc;
    int s = (int)(bs % Sc);
    int b = (int)(bs / Sc);

    float rx = ref[((long long)b * Sc + s) * 2 + 0];
    float ry = ref[((long long)b * Sc + s) * 2 + 1];
    const float* offp = off + ((((long long)b * Sc + s) * NHc + h) * (Lc * Pc * 2));
    const float* awp  = aw  + (((long long)b * Sc + s) * NHc + h) * (Lc * Pc);

    float acc = 0.0f;
    for (int lvl = 0; lvl < Lc; ++lvl) {
        const int H = d_H[lvl], W = d_W[lvl], s0 = d_S0[lvl];
        const float* vl = v + ((long long)b * Sc + s0) * Dc + h * HDc + lane;
        const float fW = (float)W, fH = (float)H;
        #pragma unroll
        for (int p = 0; p < Pc; ++p) {
            float ox = offp[(lvl * Pc + p) * 2 + 0];
            float oy = offp[(lvl * Pc + p) * 2 + 1];
            float w  = awp[lvl * Pc + p];
            float xf = (rx + ox / fW) * fW - 0.5f;
            float yf = (ry + oy / fH) * fH - 0.5f;
            float x0 = floorf(xf), y0 = floorf(yf);
            float fx = xf - x0, fy = yf - y0;
            int x0i = (int)x0, y0i = (int)y0;
            float sample = 0.0f;
            #pragma unroll
            for (int cy = 0; cy < 2; ++cy) {
                #pragma unroll
                for (int cx = 0; cx < 2; ++cx) {
                    int xi = x0i + cx, yi = y0i + cy;
                    float wgt = (cx ? fx : 1.0f - fx) * (cy ? fy : 1.0f - fy);
                    if (xi >= 0 && xi < W && yi >= 0 && yi < H)
                        sample += wgt * vl[(long long)(yi * W + xi) * Dc];
                }
            }
            acc += w * sample;
        }
    }
    out[((long long)b * Sc + s) * Dc + h * HDc + lane] = acc;
}

// =================== out = LayerNorm(a + b) * g + beta   (block = one row, 256 thr) ===================
__global__ __launch_bounds__(256) void add_layernorm_kernel(
    const float* __restrict__ a, const float* __restrict__ b,
    const float* __restrict__ g, const float* __restrict__ beta,
    float* __restrict__ o)
{
    __shared__ float red[256];
    const long long row = blockIdx.x;
    const int t = threadIdx.x; // D == 256
    float x = a[row * Dc + t] + b[row * Dc + t];

    red[t] = x;
    __syncthreads();
    for (int s = 128; s > 0; s >>= 1) { if (t < s) red[t] += red[t + s]; __syncthreads(); }
    float mean = red[0] * (1.0f / Dc);
    __syncthreads();

    float d = x - mean;
    red[t] = d * d;
    __syncthreads();
    for (int s = 128; s > 0; s >>= 1) { if (t < s) red[t] += red[t + s]; __syncthreads(); }
    float var = red[0] * (1.0f / Dc);
    float r = rsqrtf(var + EPSc);
    o[row * Dc + t] = d * r * g[t] + beta[t];
}

// =================== workspace layout (floats) ===================
constexpr long long OFF_REF  = 0;                               // B*S*2
constexpr long long OFF_Q    = 65536;                           // M*256
constexpr long long OFF_V    = OFF_Q    + MROWS * 256;          // M*256
constexpr long long OFF_OFFS = OFF_V    + MROWS * 256;          // M*256
constexpr long long OFF_AW   = OFF_OFFS + MROWS * 256;          // M*128
constexpr long long OFF_SAMP = OFF_AW   + MROWS * 128;          // M*256
constexpr long long OFF_X    = OFF_SAMP + MROWS * 256;          // M*256
constexpr long long OFF_H    = OFF_Q;   // alias: q/v/off/aw/samp dead by FFN stage

extern "C" void kernel_launch(void* const* d_in, const int* in_sizes, int n_in,
                              void* d_out, int out_size, void* d_ws, size_t ws_size,
                              hipStream_t stream) {
    const float* query  = (const float*)d_in[0];
    const float* qpos   = (const float*)d_in[1];
    const float* vr     = (const float*)d_in[2];
    // d_in[3] spatial_shapes / d_in[4] level_start_index: module constants, hardcoded
    const float* W_off  = (const float*)d_in[5];
    const float* b_off  = (const float*)d_in[6];
    const float* W_attn = (const float*)d_in[7];
    const float* b_attn = (const float*)d_in[8];
    const float* W_val  = (const float*)d_in[9];
    const float* b_val  = (const float*)d_in[10];
    const float* W_out  = (const float*)d_in[11];
    const float* b_out  = (const float*)d_in[12];
    const float* ln1_g  = (const float*)d_in[13];
    const float* ln1_b  = (const float*)d_in[14];
    const float* W1     = (const float*)d_in[15];
    const float* b1     = (const float*)d_in[16];
    const float* W2     = (const float*)d_in[17];
    const float* b2     = (const float*)d_in[18];
    const float* ln2_g  = (const float*)d_in[19];
    const float* ln2_b  = (const float*)d_in[20];

    float* out = (float*)d_out;
    float* ws  = (float*)d_ws;
    float* ref  = ws + OFF_REF;
    float* q    = ws + OFF_Q;
    float* v    = ws + OFF_V;
    float* offs = ws + OFF_OFFS;
    float* aw   = ws + OFF_AW;
    float* samp = ws + OFF_SAMP;
    float* x    = ws + OFF_X;
    float* hbuf = ws + OFF_H;

    const int M = (int)MROWS;
    const long long nQD = MROWS * 256;

    // 1) reference points
    {
        long long n = MROWS;
        ref_points_kernel<<<(unsigned)((n + 255) / 256), 256, 0, stream>>>(vr, ref);
    }
    // 2) q = query + query_pos
    add_kernel<<<(unsigned)((nQD + 255) / 256), 256, 0, stream>>>(query, qpos, q, nQD);

    dim3 gemm_blk(256);
    const unsigned mtiles = (unsigned)((M + MT - 1) / MT);

    // 3) v = query @ W_val + b_val         (M x 256 x 256)
    gemm_wmma_f32<<<dim3(mtiles, 256 / NT), gemm_blk, 0, stream>>>(query, W_val, b_val, v,    M, 256, 256, 0);
    // 4) off = q @ W_off + b_off           (M x 256 x 256)
    gemm_wmma_f32<<<dim3(mtiles, 256 / NT), gemm_blk, 0, stream>>>(q,     W_off, b_off, offs, M, 256, 256, 0);
    // 5) aw = softmax(q @ W_attn + b_attn) (M x 128 x 256)
    gemm_wmma_f32<<<dim3(mtiles, 128 / NT), gemm_blk, 0, stream>>>(q,     W_attn, b_attn, aw, M, 128, 256, 0);
    {
        long long heads = MROWS * NHc;
        softmax16_kernel<<<(unsigned)((heads + 255) / 256), 256, 0, stream>>>(aw, heads);
    }
    // 6) deformable sampling -> samp (B,S,D)
    {
        long long waves = MROWS * NHc;                // one wave32 per (b,s,h)
        long long thr   = waves * 32;
        deform_attn_kernel<<<(unsigned)((thr + 255) / 256), 256, 0, stream>>>(v, offs, aw, ref, samp);
    }
    // 7) attn_out = samp @ W_out + b_out  -> reuse q region (q is dead)
    float* attn_out = q;
    gemm_wmma_f32<<<dim3(mtiles, 256 / NT), gemm_blk, 0, stream>>>(samp, W_out, b_out, attn_out, M, 256, 256, 0);
    // 8) x = LN(query + attn_out)
    add_layernorm_kernel<<<(unsigned)MROWS, 256, 0, stream>>>(query, attn_out, ln1_g, ln1_b, x);
    // 9) h = relu(x @ W1 + b1)             (M x 1024 x 256), h aliases dead buffers
    gemm_wmma_f32<<<dim3(mtiles, 1024 / NT), gemm_blk, 0, stream>>>(x, W1, b1, hbuf, M, 1024, 256, 1);
    // 10) d_out = h @ W2 + b2              (M x 256 x 1024)
    gemm_wmma_f32<<<dim3(mtiles, 256 / NT), gemm_blk, 0, stream>>>(hbuf, W2, b2, out, M, 256, 1024, 0);
    // 11) d_out = LN(x + d_out)  (in-place safe: row read fully before write)
    add_layernorm_kernel<<<(unsigned)MROWS, 256, 0, stream>>>(x, out, ln2_g, ln2_b, out);
}